// MyModel_54649163875057
// MI455X (gfx1250) — compile-verified
//
#include <hip/hip_runtime.h>

typedef __attribute__((ext_vector_type(2))) float v2f;
typedef __attribute__((ext_vector_type(8))) float v8f;

#define HH     64
#define WW     64
#define CIN    128
#define COUT   128
#define KKF    9          // K*K
#define NPOS   128        // spatial positions per workgroup (2 rows of 64)
#define CHUNK  4          // cin per K-chunk
#define KC     (CHUNK*KKF)   // 36 K values per chunk
#define KSTEPS (KC/4)        // 9 wmma K-steps per chunk
#define NCHUNK (CIN/CHUNK)   // 32

__launch_bounds__(256)
__global__ void deform_conv_wmma_kernel(const float* __restrict__ x,
                                        const float* __restrict__ offset,
                                        const float* __restrict__ weight,
                                        float* __restrict__ out) {
    // Pair-interleaved tiles: element [kp][j] holds k = 2*kp (.x) and k = 2*kp+1 (.y)
    __shared__ v2f    wP[(KC/2) * COUT];     // 18*128*8B = 18.0 KB  (A: weights)
    __shared__ v2f    colP[(KC/2) * NPOS];   // 18*128*8B = 18.0 KB  (B: sampled cols)
    __shared__ float4 metaW[KKF * NPOS];     // 9*128*16B = 18.0 KB  (mask-folded bilinear wts)
    __shared__ int    metaC[KKF * NPOS];     // 9*128*4B  =  4.5 KB  (packed clamped y0|x0<<8)

    const int tid  = threadIdx.x;
    const int lane = tid & 31;
    const int wv   = tid >> 5;       // wave 0..7 -> cout tile
    const int half = lane >> 4;      // WMMA lane-half
    const int lm   = lane & 15;
    const int m0   = wv * 16;

    const int n  = blockIdx.x >> 5;          // image 0..7
    const int p0 = (blockIdx.x & 31) * NPOS; // spatial tile base

    const float* offn = offset + (size_t)n * (2 * KKF) * (HH * WW);
    const float* xn   = x      + (size_t)n * CIN * (HH * WW);

    // ---- Phase A: sampling metadata for this tile (once) ----
    for (int e = tid; e < KKF * NPOS; e += 256) {
        int kk = e >> 7;                 // 0..8
        int t  = e & (NPOS - 1);
        int p  = p0 + t;
        int ho = p >> 6, wo = p & 63;
        float dy = offn[((kk * 2 + 0) * HH + ho) * WW + wo];
        float dx = offn[((kk * 2 + 1) * HH + ho) * WW + wo];
        float py = (float)(ho - 1 + (kk / 3)) + dy;   // STRIDE=1, PAD=1, DIL=1
        float px = (float)(wo - 1 + (kk % 3)) + dx;
        float fy = floorf(py), fx = floorf(px);
        int   y0 = (int)fy,    x0 = (int)fx;
        float wy1 = py - fy,   wx1 = px - fx;
        float wy0 = 1.0f - wy1, wx0 = 1.0f - wx1;
        float my0 = (y0 >= 0     && y0     < HH) ? 1.0f : 0.0f;
        float my1 = (y0 + 1 >= 0 && y0 + 1 < HH) ? 1.0f : 0.0f;
        float mx0 = (x0 >= 0     && x0     < WW) ? 1.0f : 0.0f;
        float mx1 = (x0 + 1 >= 0 && x0 + 1 < WW) ? 1.0f : 0.0f;
        float4 w4;
        w4.x = wy0 * wx0 * my0 * mx0;   // mask folded into weights ->
        w4.y = wy0 * wx1 * my0 * mx1;   // gather phase is branch-free FMA
        w4.z = wy1 * wx0 * my1 * mx0;
        w4.w = wy1 * wx1 * my1 * mx1;
        metaW[e] = w4;
        int y0c = min(max(y0, 0), HH - 1);
        int x0c = min(max(x0, 0), WW - 1);
        metaC[e] = y0c | (x0c << 8);
    }
    __syncthreads();

    v8f acc[8];
    #pragma unroll
    for (int i = 0; i < 8; ++i) acc[i] = (v8f){0.f,0.f,0.f,0.f,0.f,0.f,0.f,0.f};

    float* wPf   = (float*)wP;
    float* colPf = (float*)colP;

    for (int cc = 0; cc < NCHUNK; ++cc) {
        // ---- Weight chunk -> LDS (coalesced float4 reads; chunk is 36 contiguous f32/row) ----
        for (int e4 = tid; e4 < COUT * KSTEPS; e4 += 256) {   // 1152 float4 loads
            int co = e4 / KSTEPS;
            int q  = e4 - co * KSTEPS;
            float4 w4 = ((const float4*)(weight + (size_t)co * (CIN * KKF) + cc * KC))[q];
            int kb = q * 4;
            wPf[(( kb      >> 1) * COUT + co) * 2 + 0] = w4.x;  // k = kb
            wPf[(( kb      >> 1) * COUT + co) * 2 + 1] = w4.y;  // k = kb+1
            wPf[(((kb + 2) >> 1) * COUT + co) * 2 + 0] = w4.z;  // k = kb+2
            wPf[(((kb + 2) >> 1) * COUT + co) * 2 + 1] = w4.w;  // k = kb+3
        }

        // ---- Deformable gather: build 36 x 128 col tile in LDS ----
        const float* xcc = xn + (size_t)cc * CHUNK * (HH * WW);
        #pragma unroll
        for (int i = 0; i < (KC * NPOS) / 256; ++i) {          // 18 samples/thread
            int e    = tid + i * 256;
            int kloc = e >> 7;                                 // 0..35
            int t    = e & (NPOS - 1);
            int ci   = kloc / KKF;
            int kk   = kloc - ci * KKF;
            float4 w4 = metaW[kk * NPOS + t];                  // ds_load_b128
            int pc    = metaC[kk * NPOS + t];
            int y0 = pc & 255, x0 = (pc >> 8) & 255;
            int y1 = min(y0 + 1, HH - 1), x1 = min(x0 + 1, WW - 1);
            const float* pl = xcc + ci * (HH * WW);
            float v = w4.x * pl[y0 * WW + x0] + w4.y * pl[y0 * WW + x1]
                    + w4.z * pl[y1 * WW + x0] + w4.w * pl[y1 * WW + x1];
            colPf[((kloc >> 1) * NPOS + t) * 2 + (kloc & 1)] = v;
        }
        __syncthreads();

        // ---- fp32 WMMA over the chunk: 9 K-steps x 8 N-tiles ----
        #pragma unroll
        for (int ks = 0; ks < KSTEPS; ++ks) {
            int kp = ks * 2 + half;                            // pair row for this lane-half
            v2f a = wP[kp * COUT + m0 + lm];                   // single ds_load_b64
            #pragma unroll
            for (int nt = 0; nt < 8; ++nt) {
                v2f b = colP[kp * NPOS + nt * 16 + lm];        // single ds_load_b64
                acc[nt] = __builtin_amdgcn_wmma_f32_16x16x4_f32(
                    /*neg_a=*/false, a, /*neg_b=*/false, b,
                    /*c_mod=*/(short)0, acc[nt],
                    /*reuse_a=*/false, /*reuse_b=*/false);
            }
        }
        __syncthreads();
    }

    // ---- Write 16(cout) x 128(pos) strip per wave ----
    // D layout: VGPR r, lanes 0-15 -> M=r, lanes 16-31 -> M=r+8; N = lane%16.
    float* outp = out + ((size_t)(n * COUT + m0 + half * 8)) * (HH * WW) + p0 + lm;
    #pragma unroll
    for (int nt = 0; nt < 8; ++nt) {
        #pragma unroll
        for (int r = 0; r < 8; ++r) {
            outp[(size_t)r * (HH * WW) + nt * 16] = acc[nt][r];
        }
    }
}

extern "C" void kernel_launch(void* const* d_in, const int* in_sizes, int n_in,
                              void* d_out, int out_size, void* d_ws, size_t ws_size,
                              hipStream_t stream) {
    const float* x      = (const float*)d_in[0];  // (8,128,64,64) f32
    const float* offset = (const float*)d_in[1];  // (8,18,64,64)  f32
    const float* weight = (const float*)d_in[2];  // (128,128,3,3) f32
    float*       out    = (float*)d_out;          // (8,128,64,64) f32

    dim3 grid(8 * ((HH * WW) / NPOS));            // 8 images x 32 spatial tiles = 256
    dim3 block(256);                              // 8 wave32
    deform_conv_wmma_kernel<<<grid, block, 0, stream>>>(x, offset, weight, out);
}